// AttentionScorer_9028021256255
// MI455X (gfx1250) — compile-verified
//
#include <hip/hip_runtime.h>
#include <hip/hip_bf16.h>
#include <math.h>

typedef __attribute__((ext_vector_type(2))) float v2f;
typedef __attribute__((ext_vector_type(8))) float v8f;

#define NB   2
#define CTX  1024
#define QRS  256
#define DD   256
#define CTILE 8

__device__ __forceinline__ float fast_tanh(float x) {
#if __has_builtin(__builtin_amdgcn_tanhf)
    return __builtin_amdgcn_tanhf(x);   // v_tanh_f32 (gfx1250 TRANS op)
#else
    return tanhf(x);
#endif
}

// out[b, m, n] = sum_k A[b, m, k] * W[n, k]  (+ bias[n] if BIAS)
// One wave computes one 16x16 output tile via V_WMMA_F32_16X16X4_F32.
// A frag (16x4 f32): lane m in 0..15, group g = lane>=16; VGPR j holds A[m, k0 + j + 2g].
// B frag (4x16 f32): lane n = lane&15; VGPR j holds B[k0 + j + 2g, n] = W[n, k0 + j + 2g].
// C/D (16x16 f32): VGPR i holds row (i + 8g), col = lane&15.
template <int M, bool BIAS>
__global__ __launch_bounds__(256) void gemm_nt_wmma(
    const float* __restrict__ A, const float* __restrict__ W,
    const float* __restrict__ bias, float* __restrict__ out) {
    const int wid  = (blockIdx.x * blockDim.x + threadIdx.x) >> 5;
    const int lane = threadIdx.x & 31;
    constexpr int tilesN = DD / 16;                  // 16
    constexpr int tilesPerBatch = (M / 16) * tilesN;
    const int b    = wid / tilesPerBatch;
    const int r    = wid % tilesPerBatch;
    const int row0 = (r / tilesN) * 16;
    const int col0 = (r % tilesN) * 16;
    const int m    = lane & 15;
    const int g    = lane >> 4;

    const float* arow = A + ((size_t)b * M + row0 + m) * DD + 2 * g;
    const float* wrow = W + (size_t)(col0 + m) * DD + 2 * g;

    v8f acc = {};
#pragma unroll 8
    for (int k0 = 0; k0 < DD; k0 += 4) {
        v2f av = *(const v2f*)(arow + k0);
        v2f bv = *(const v2f*)(wrow + k0);
        acc = __builtin_amdgcn_wmma_f32_16x16x4_f32(
            /*neg_a=*/false, av, /*neg_b=*/false, bv,
            /*c_mod=*/(short)0, acc, /*reuse_a=*/false, /*reuse_b=*/false);
    }

    float bval = BIAS ? bias[col0 + m] : 0.f;
    float* orow = out + ((size_t)b * M + row0 + 8 * g) * DD + col0 + m;
#pragma unroll
    for (int i = 0; i < 8; ++i)
        orow[(size_t)i * DD] = acc[i] + bval;
}

// scores[b, c, q] = sum_e v[e] * tanh(wq[b,c,e] + uh[b,q,e])
// Block = (batch b, 8 context rows); thread t = query q. wq/v addresses are
// uniform per block/loop -> scalar (SMEM) loads; uh is a per-lane float4 stream.
__global__ __launch_bounds__(256) void score_tanh_kernel(
    const float* __restrict__ wq, const float* __restrict__ uh,
    const float* __restrict__ v, float* __restrict__ out) {
    const int b  = blockIdx.x / (CTX / CTILE);
    const int c0 = (blockIdx.x % (CTX / CTILE)) * CTILE;
    const int q  = threadIdx.x;

    const float* uhp = uh + ((size_t)b * QRS + q) * DD;
    const float* wqp = wq + ((size_t)b * CTX + c0) * DD;

    float acc[CTILE];
#pragma unroll
    for (int i = 0; i < CTILE; ++i) acc[i] = 0.f;

#pragma unroll 2
    for (int e = 0; e < DD; e += 4) {
        const float4 u  = *(const float4*)(uhp + e);
        const float4 vv = *(const float4*)(v + e);
#pragma unroll
        for (int i = 0; i < 4; ++i) {
            const float ue = (&u.x)[i];
            const float ve = (&vv.x)[i];
#pragma unroll
            for (int ct = 0; ct < CTILE; ++ct)
                acc[ct] += ve * fast_tanh(wqp[(size_t)ct * DD + e + i] + ue);
        }
    }

    float* op = out + ((size_t)b * CTX + c0) * QRS + q;   // flat (B, CTX, QRS)
#pragma unroll
    for (int ct = 0; ct < CTILE; ++ct)
        op[(size_t)ct * QRS] = acc[ct];
}

extern "C" void kernel_launch(void* const* d_in, const int* in_sizes, int n_in,
                              void* d_out, int out_size, void* d_ws, size_t ws_size,
                              hipStream_t stream) {
    const float* context = (const float*)d_in[0];  // (2, 1024, 256)
    const float* queries = (const float*)d_in[1];  // (2, 256, 256)
    const float* Wc      = (const float*)d_in[2];  // (256, 256)
    const float* Wq      = (const float*)d_in[3];  // (256, 256)
    const float* bq      = (const float*)d_in[4];  // (256,)
    const float* v       = (const float*)d_in[5];  // (1, 256)
    float* out = (float*)d_out;                    // (2, 1024, 256) flat

    float* wq = (float*)d_ws;                         // NB*CTX*DD floats = 2 MB
    float* uh = wq + (size_t)NB * CTX * DD;           // NB*QRS*DD floats = 0.5 MB

    // Phase 1: wq = context @ Wc^T   (2048 waves -> 256 blocks of 8 waves)
    {
        int waves  = NB * (CTX / 16) * (DD / 16);
        int blocks = waves * 32 / 256;
        gemm_nt_wmma<CTX, false><<<blocks, 256, 0, stream>>>(context, Wc, nullptr, wq);
    }
    // Phase 2: uh = queries @ Wq^T + bq   (512 waves -> 64 blocks)
    {
        int waves  = NB * (QRS / 16) * (DD / 16);
        int blocks = waves * 32 / 256;
        gemm_nt_wmma<QRS, true><<<blocks, 256, 0, stream>>>(queries, Wq, bq, uh);
    }
    // Phase 3: tanh-reduce (dominant cost: 134M v_tanh_f32)
    score_tanh_kernel<<<NB * CTX / CTILE, 256, 0, stream>>>(wq, uh, v, out);
}